// xLSTM_Block_14903536517287
// MI455X (gfx1250) — compile-verified
//
#include <hip/hip_runtime.h>
#include <math.h>

// Problem constants (B=4, S=2048, IN=256, H=M=128)
#define NTOK   8192
#define IND    256
#define HD     128
#define CELEM  16384   // 128*128 per-token c block

// d_out layout: h [NTOK*HD] | c [NTOK*CELEM] | n [NTOK*HD]
static constexpr size_t H_OFF = 0;
static constexpr size_t C_OFF = (size_t)NTOK * HD;                 // 1,048,576
static constexpr size_t N_OFF = C_OFF + (size_t)NTOK * CELEM;      // + 134,217,728

// Staging layout inside each token's c block (overwritten by finalize):
//  [0..127]=q  [128..255]=k(scaled)  [256..383]=v  [384..511]=o(sigmoid)
//  [512..639]=cq (= c0 . q)  [640]=i_t  [641]=f_t
#define STG_Q   0
#define STG_K   128
#define STG_V   256
#define STG_O   384
#define STG_CQ  512
#define STG_I   640
#define STG_F   641

typedef __attribute__((ext_vector_type(2))) float v2f;
typedef __attribute__((ext_vector_type(4))) float v4f;
typedef __attribute__((ext_vector_type(8))) float v8f;

// ---------------------------------------------------------------------------
// Kernel 1: projections  X[8192,256] @ {w_q,w_k,w_v,w_o}[256,128] (+bias,act)
// One wave = one 16x16 output tile. fp32 WMMA, K stepped by 4.
// grid = (NTOK/16, 4), block = 256 (8 waves); colGroup = by*8+wave in [0,32)
// ---------------------------------------------------------------------------
__global__ void __launch_bounds__(256) proj_wmma_kernel(
    const float* __restrict__ x,
    const float* __restrict__ wq, const float* __restrict__ bq,
    const float* __restrict__ wk, const float* __restrict__ bk,
    const float* __restrict__ wv, const float* __restrict__ bv,
    const float* __restrict__ wo, const float* __restrict__ bo,
    float* __restrict__ out)
{
    const int wave  = threadIdx.x >> 5;
    const int lane  = threadIdx.x & 31;
    const int cg    = blockIdx.y * 8 + wave;   // 0..31
    const int mat   = cg >> 3;                 // 0:q 1:k 2:v 3:o
    const int colBase = (cg & 7) * 16;
    const int tokBase = blockIdx.x * 16;

    const float* W; const float* Bias; int stgOff;
    switch (mat) {
        case 0:  W = wq; Bias = bq; stgOff = STG_Q; break;
        case 1:  W = wk; Bias = bk; stgOff = STG_K; break;
        case 2:  W = wv; Bias = bv; stgOff = STG_V; break;
        default: W = wo; Bias = bo; stgOff = STG_O; break;
    }

    const int r16   = lane & 15;
    const int khalf = lane >> 4;
    const int col   = colBase + r16;

    const float* arow = x + (size_t)(tokBase + r16) * IND + 2 * khalf;
    const float* bcol = W + (size_t)(2 * khalf) * HD + col;

    v8f acc = {0.f,0.f,0.f,0.f,0.f,0.f,0.f,0.f};
    #pragma unroll 8
    for (int kk = 0; kk < IND; kk += 4) {
        v2f a, b;
        a.x = arow[kk];
        a.y = arow[kk + 1];
        b.x = bcol[(size_t)kk * HD];
        b.y = bcol[(size_t)(kk + 1) * HD];
        acc = __builtin_amdgcn_wmma_f32_16x16x4_f32(
            false, a, false, b, (short)0, acc, false, false);
    }

    const float bias = Bias[col];
    float* stg = out + C_OFF;
    #pragma unroll
    for (int r = 0; r < 8; ++r) {
        float v = acc[r] + bias;
        if (mat == 1)      v *= 0.08838834764831845f;        // 1/sqrt(128)
        else if (mat == 3) v = 1.0f / (1.0f + expf(-v));     // sigmoid(o)
        const int tok = tokBase + r + 8 * khalf;
        stg[(size_t)tok * CELEM + stgOff + col] = v;
    }
}

// ---------------------------------------------------------------------------
// Kernel 2: scalar gates i_t = exp(x.w_i+b_i), f_t = sigmoid(x.w_f+b_f)
// one wave per token, lane-strided coalesced loads + wave32 xor reduction
// ---------------------------------------------------------------------------
__global__ void __launch_bounds__(256) gates_kernel(
    const float* __restrict__ x,
    const float* __restrict__ wi, const float* __restrict__ bi,
    const float* __restrict__ wf, const float* __restrict__ bf,
    float* __restrict__ out)
{
    const int wave = threadIdx.x >> 5;
    const int lane = threadIdx.x & 31;
    const int n    = blockIdx.x * 8 + wave;
    const float* row = x + (size_t)n * IND;

    float si = 0.f, sf = 0.f;
    #pragma unroll
    for (int k = lane; k < IND; k += 32) {
        const float xv = row[k];
        si += xv * wi[k];
        sf += xv * wf[k];
    }
    #pragma unroll
    for (int off = 16; off > 0; off >>= 1) {
        si += __shfl_xor(si, off, 32);
        sf += __shfl_xor(sf, off, 32);
    }
    if (lane == 0) {
        float* stg = out + C_OFF + (size_t)n * CELEM;
        stg[STG_I] = expf(si + bi[0]);
        stg[STG_F] = 1.0f / (1.0f + expf(-(sf + bf[0])));
    }
}

// ---------------------------------------------------------------------------
// Kernel 3: cq = Q @ c0^T   (cq[n,m] = sum_k c0[m,k] * q[n,k]), K=128
// grid = NTOK/16, block = 256 (8 waves, one 16-col group each)
// ---------------------------------------------------------------------------
__global__ void __launch_bounds__(256) cq_wmma_kernel(
    const float* __restrict__ c0, float* __restrict__ out)
{
    const int wave  = threadIdx.x >> 5;
    const int lane  = threadIdx.x & 31;
    const int colBase = wave * 16;
    const int tokBase = blockIdx.x * 16;

    const int r16   = lane & 15;
    const int khalf = lane >> 4;
    const int col   = colBase + r16;

    const float* stg  = out + C_OFF;
    const float* arow = stg + (size_t)(tokBase + r16) * CELEM + STG_Q + 2 * khalf;
    const float* brow = c0 + (size_t)col * HD + 2 * khalf;   // B[k][col] = c0[col][k]

    v8f acc = {0.f,0.f,0.f,0.f,0.f,0.f,0.f,0.f};
    #pragma unroll 8
    for (int kk = 0; kk < HD; kk += 4) {
        v2f a, b;
        a.x = arow[kk];
        a.y = arow[kk + 1];
        b.x = brow[kk];
        b.y = brow[kk + 1];
        acc = __builtin_amdgcn_wmma_f32_16x16x4_f32(
            false, a, false, b, (short)0, acc, false, false);
    }

    float* stgw = out + C_OFF;
    #pragma unroll
    for (int r = 0; r < 8; ++r) {
        const int tok = tokBase + r + 8 * khalf;
        stgw[(size_t)tok * CELEM + STG_CQ + col] = acc[r];
    }
}

// ---------------------------------------------------------------------------
// Kernel 4: finalize. One block per token: read staging -> LDS, then write
// h[n], n[n], and the 64KB c block with coalesced non-temporal 128-bit stores.
// The 512MB c stream is write-once/never-reread -> NT hint keeps c0/x/staging
// resident in the 192MB L2 instead of being evicted by the stream.
// ---------------------------------------------------------------------------
__global__ void __launch_bounds__(256) finalize_kernel(
    const float* __restrict__ c0, const float* __restrict__ n0,
    float* __restrict__ out)
{
    __shared__ __align__(16) float c0s[HD * HD];   // 64 KB
    __shared__ __align__(16) float ks[HD];
    __shared__ __align__(16) float vs[HD];
    __shared__ float qs[HD], os[HD], cqs[HD];
    __shared__ float redA[HD], redB[HD];
    __shared__ float s_if[2];

    const int n = blockIdx.x;
    const int t = threadIdx.x;

    // stage c0 into LDS (128-bit, coalesced, regular-temporal: L2-resident)
    const v4f* c0g4 = (const v4f*)c0;
    v4f*       c0s4 = (v4f*)c0s;
    #pragma unroll
    for (int it = 0; it < 16; ++it)
        c0s4[t + it * 256] = c0g4[t + it * 256];

    // pull per-token staging out of the c block before we overwrite it
    const float* stg = out + C_OFF + (size_t)n * CELEM;
    if (t < HD) {
        qs[t]  = stg[STG_Q  + t];
        ks[t]  = stg[STG_K  + t];
        vs[t]  = stg[STG_V  + t];
        os[t]  = stg[STG_O  + t];
        cqs[t] = stg[STG_CQ + t];
    }
    if (t == 0) { s_if[0] = stg[STG_I]; s_if[1] = stg[STG_F]; }
    __syncthreads();

    // reductions: kq = k.q, n0q = n0.q
    if (t < HD) {
        const float qv = qs[t];
        redA[t] = ks[t] * qv;
        redB[t] = n0[t] * qv;
    }
    __syncthreads();
    for (int s = 64; s > 0; s >>= 1) {
        if (t < s) { redA[t] += redA[t + s]; redB[t] += redB[t + s]; }
        __syncthreads();
    }
    const float kq  = redA[0];
    const float n0q = redB[0];
    const float ii  = s_if[0];
    const float fi  = s_if[1];

    const float nq    = fi * n0q + ii * kq;
    const float denom = fmaxf(fabsf(nq), 1.0f);
    const float rden  = 1.0f / denom;

    if (t < HD) {
        // h = o * (f*(c0.q) + i*v*(k.q)) / denom
        const float hv = os[t] * (fi * cqs[t] + ii * vs[t] * kq) * rden;
        __builtin_nontemporal_store(hv, &out[H_OFF + (size_t)n * HD + t]);
        // n = f*n0 + i*k
        const float nv = fi * n0[t] + ii * ks[t];
        __builtin_nontemporal_store(nv, &out[N_OFF + (size_t)n * HD + t]);
    }

    // c[n,m,j] = f*c0[m,j] + (i*v[m]) * k[j]  -- the 512MB NT stream
    const int j4 = t & 31;                 // fixed 128-bit column per thread
    const v4f kv = ((const v4f*)ks)[j4];
    v4f* cdst = (v4f*)(out + C_OFF + (size_t)n * CELEM);
    #pragma unroll
    for (int it = 0; it < 16; ++it) {
        const int m  = (t >> 5) + it * 8;
        const float sc = ii * vs[m];
        const v4f c0v = c0s4[m * 32 + j4];
        v4f r;
        r.x = fi * c0v.x + sc * kv.x;
        r.y = fi * c0v.y + sc * kv.y;
        r.z = fi * c0v.z + sc * kv.z;
        r.w = fi * c0v.w + sc * kv.w;
        __builtin_nontemporal_store(r, &cdst[m * 32 + j4]);
    }
}

// ---------------------------------------------------------------------------
extern "C" void kernel_launch(void* const* d_in, const int* in_sizes, int n_in,
                              void* d_out, int out_size, void* d_ws, size_t ws_size,
                              hipStream_t stream) {
    const float* x  = (const float*)d_in[0];
    const float* c0 = (const float*)d_in[1];
    const float* n0 = (const float*)d_in[2];
    const float* wq = (const float*)d_in[3];
    const float* bq = (const float*)d_in[4];
    const float* wk = (const float*)d_in[5];
    const float* bk = (const float*)d_in[6];
    const float* wv = (const float*)d_in[7];
    const float* bv = (const float*)d_in[8];
    const float* wi = (const float*)d_in[9];
    const float* bi = (const float*)d_in[10];
    const float* wf = (const float*)d_in[11];
    const float* bf = (const float*)d_in[12];
    const float* wo = (const float*)d_in[13];
    const float* bo = (const float*)d_in[14];
    float* out = (float*)d_out;
    (void)in_sizes; (void)n_in; (void)out_size; (void)d_ws; (void)ws_size;

    // 1) projections q,k,v,o (WMMA fp32) -> staged in c region
    proj_wmma_kernel<<<dim3(NTOK / 16, 4), 256, 0, stream>>>(
        x, wq, bq, wk, bk, wv, bv, wo, bo, out);
    // 2) scalar gates i_t, f_t
    gates_kernel<<<NTOK / 8, 256, 0, stream>>>(x, wi, bi, wf, bf, out);
    // 3) cq = Q @ c0^T (WMMA fp32)
    cq_wmma_kernel<<<NTOK / 16, 256, 0, stream>>>(c0, out);
    // 4) finalize: h, n, and the 512MB non-temporal c stream
    finalize_kernel<<<NTOK, 256, 0, stream>>>(c0, n0, out);
}